// DGC_4475355922586
// MI455X (gfx1250) — compile-verified
//
#include <hip/hip_runtime.h>
#include <math.h>

// ---------------- problem constants (match reference) ----------------
#define EPSILON     0.1f
#define ITERATIONS  4
#define IN_DIM      128
#define HID_DIM     64
#define OUT_DIM     64

typedef __attribute__((ext_vector_type(2))) float v2f;
typedef __attribute__((ext_vector_type(8))) float v8f;

__device__ __forceinline__ void atomic_add_f32(float* p, float v) {
    (void)__hip_atomic_fetch_add(p, v, __ATOMIC_RELAXED, __HIP_MEMORY_SCOPE_AGENT);
}

// Async global->LDS copy of 16 bytes per lane (gfx1250 ASYNCcnt path).
__device__ __forceinline__ void async_copy_b128(const float* gsrc, float* lds_dst) {
    unsigned lds_off = (unsigned)(size_t)lds_dst;    // low 32 bits of generic = LDS offset
    asm volatile("global_load_async_to_lds_b128 %0, %1, off"
                 :: "v"(lds_off), "v"(gsrc) : "memory");
}
__device__ __forceinline__ void async_wait0() {
    asm volatile("s_wait_asynccnt 0x0" ::: "memory");
}

// =====================================================================
// Embed GEMM: H[N,64] = X[N,128] @ W[128,64] + b   (fp32 WMMA 16x16x4)
// One wave -> 16 rows x 64 cols (4 N-tiles share the A fragment).
// Block = 256 threads = 8 waves = 128 rows.
// B fragments are pre-paired in LDS: sWp[p*64+n] = {W[2p][n], W[2p+1][n]}
// so each fragment is one aligned ds_load_b64 straight into a VGPR pair.
// =====================================================================
__global__ __launch_bounds__(256) void gemm_embed_kernel(
    const float* __restrict__ X, const float* __restrict__ W,
    const float* __restrict__ bias, float* __restrict__ H, int nNodes)
{
    __shared__ float sraw[IN_DIM * HID_DIM];             // 32 KB (async-staged raw W)
    __shared__ v2f   sWp[(IN_DIM / 2) * HID_DIM];        // 32 KB (pair-major W)

    for (int i = threadIdx.x; i < (IN_DIM * HID_DIM) / 4; i += 256)
        async_copy_b128(W + i * 4, &sraw[i * 4]);
    async_wait0();
    __syncthreads();

    // repack raw -> pair-major (one time, 16 iters/thread)
    for (int i = threadIdx.x; i < (IN_DIM / 2) * HID_DIM; i += 256) {
        const int p = i >> 6;                            // pair index (HID_DIM == 64)
        const int n = i & 63;
        v2f t;
        t.x = sraw[(2 * p)     * HID_DIM + n];
        t.y = sraw[(2 * p + 1) * HID_DIM + n];
        sWp[i] = t;
    }
    __syncthreads();

    const int lane    = threadIdx.x & 31;
    const int wave    = threadIdx.x >> 5;
    const int m       = lane & 15;                       // row-in-tile / col-in-tile
    const int kk      = (lane >> 4) << 1;                // 0 or 2 (K sub-pair)
    const int phalf   = lane >> 4;                       // kk/2
    const int rowBase = blockIdx.x * 128 + wave * 16;

    int row  = rowBase + m;
    int rowc = row < nNodes ? row : nNodes - 1;          // clamp: keep EXEC all-1s for WMMA
    const float* xr = X + (long)rowc * IN_DIM;

    v8f acc[4] = {};
    for (int k = 0; k < IN_DIM; k += 4) {
        // A fragment: adjacent pair, streamed once -> non-temporal b64 load
        v2f a = __builtin_nontemporal_load((const v2f*)(xr + k + kk));
        const int pbase = ((k >> 1) + phalf) * HID_DIM + m;
#pragma unroll
        for (int t = 0; t < 4; ++t) {
            v2f b = sWp[pbase + t * 16];                 // one ds_load_b64
            acc[t] = __builtin_amdgcn_wmma_f32_16x16x4_f32(
                false, a, false, b, (short)0, acc[t], false, false);
        }
    }

    // Epilogue: C/D layout -> lane L writes rows {rowBase + (L>=16?8:0) + r}, col t*16+(L&15)
    const int mOff = (lane >> 4) * 8;
    if (__builtin_amdgcn_readfirstlane(rowBase) + 16 <= nNodes) {
#pragma unroll
        for (int t = 0; t < 4; ++t) {
            const int col = t * 16 + m;
            const float bb = bias[col];
            float* hp = H + (long)(rowBase + mOff) * HID_DIM + col;
#pragma unroll
            for (int r = 0; r < 8; ++r)
                hp[(long)r * HID_DIM] = acc[t][r] + bb;
        }
    } else {
#pragma unroll
        for (int t = 0; t < 4; ++t) {
            const int col = t * 16 + m;
            const float bb = bias[col];
#pragma unroll
            for (int r = 0; r < 8; ++r) {
                const int outRow = rowBase + mOff + r;
                if (outRow < nNodes)
                    H[(long)outRow * HID_DIM + col] = acc[t][r] + bb;
            }
        }
    }
}

// =====================================================================
// Readout GEMM: OUT[N,64] = tanh(H[N,64]) @ W[64,64] + b
// =====================================================================
__global__ __launch_bounds__(256) void gemm_readout_kernel(
    const float* __restrict__ H, const float* __restrict__ W,
    const float* __restrict__ bias, float* __restrict__ OUT, int nNodes)
{
    __shared__ float sraw[HID_DIM * OUT_DIM];            // 16 KB
    __shared__ v2f   sWp[(HID_DIM / 2) * OUT_DIM];       // 16 KB

    for (int i = threadIdx.x; i < (HID_DIM * OUT_DIM) / 4; i += 256)
        async_copy_b128(W + i * 4, &sraw[i * 4]);
    async_wait0();
    __syncthreads();

    for (int i = threadIdx.x; i < (HID_DIM / 2) * OUT_DIM; i += 256) {
        const int p = i >> 6;                            // OUT_DIM == 64
        const int n = i & 63;
        v2f t;
        t.x = sraw[(2 * p)     * OUT_DIM + n];
        t.y = sraw[(2 * p + 1) * OUT_DIM + n];
        sWp[i] = t;
    }
    __syncthreads();

    const int lane    = threadIdx.x & 31;
    const int wave    = threadIdx.x >> 5;
    const int m       = lane & 15;
    const int kk      = (lane >> 4) << 1;
    const int phalf   = lane >> 4;
    const int rowBase = blockIdx.x * 128 + wave * 16;

    int row  = rowBase + m;
    int rowc = row < nNodes ? row : nNodes - 1;
    const float* hr = H + (long)rowc * HID_DIM;

    v8f acc[4] = {};
    for (int k = 0; k < HID_DIM; k += 4) {
        v2f raw = *(const v2f*)(hr + k + kk);
        v2f a;
        a.x = tanhf(raw.x);
        a.y = tanhf(raw.y);
        const int pbase = ((k >> 1) + phalf) * OUT_DIM + m;
#pragma unroll
        for (int t = 0; t < 4; ++t) {
            v2f b = sWp[pbase + t * 16];
            acc[t] = __builtin_amdgcn_wmma_f32_16x16x4_f32(
                false, a, false, b, (short)0, acc[t], false, false);
        }
    }

    const int mOff = (lane >> 4) * 8;
    if (__builtin_amdgcn_readfirstlane(rowBase) + 16 <= nNodes) {
#pragma unroll
        for (int t = 0; t < 4; ++t) {
            const int col = t * 16 + m;
            const float bb = bias[col];
            float* op = OUT + (long)(rowBase + mOff) * OUT_DIM + col;
#pragma unroll
            for (int r = 0; r < 8; ++r)
                op[(long)r * OUT_DIM] = acc[t][r] + bb;
        }
    } else {
#pragma unroll
        for (int t = 0; t < 4; ++t) {
            const int col = t * 16 + m;
            const float bb = bias[col];
#pragma unroll
            for (int r = 0; r < 8; ++r) {
                const int outRow = rowBase + mOff + r;
                if (outRow < nNodes)
                    OUT[(long)outRow * OUT_DIM + col] = acc[t][r] + bb;
            }
        }
    }
}

// =====================================================================
// GCN-norm precompute
// =====================================================================
__global__ void deg_init_kernel(float* __restrict__ deg, int n) {
    int i = blockIdx.x * blockDim.x + threadIdx.x;
    if (i < n) deg[i] = 2.0f;                        // self-loop weight (improved=True)
}

__global__ void deg_edges_kernel(const int* __restrict__ dst, float* __restrict__ deg, int nE) {
    int e = blockIdx.x * blockDim.x + threadIdx.x;
    if (e < nE) atomic_add_f32(&deg[dst[e]], 1.0f);
}

// In-place: deg -> dinv; also selfco[i] = 1 - eps * (2 / deg)
__global__ void norm_finalize_kernel(float* __restrict__ dinv, float* __restrict__ selfco, int n) {
    int i = blockIdx.x * blockDim.x + threadIdx.x;
    if (i < n) {
        float d = dinv[i];                           // deg >= 2, always > 0
        dinv[i]   = rsqrtf(d);
        selfco[i] = 1.0f - EPSILON * (2.0f / d);
    }
}

// coef[e] = -eps * dinv[src] * dinv[dst]
__global__ void coef_kernel(const int* __restrict__ src, const int* __restrict__ dst,
                            const float* __restrict__ dinv, float* __restrict__ coef, int nE) {
    int e = blockIdx.x * blockDim.x + threadIdx.x;
    if (e < nE) coef[e] = -EPSILON * dinv[src[e]] * dinv[dst[e]];
}

// =====================================================================
// Per-iteration kernels
// next[i,:] = h[i,:] * selfco[i]    (vectorized float4, 16 float4 per row)
// =====================================================================
__global__ void scale_rows_kernel(const float* __restrict__ h, const float* __restrict__ selfco,
                                  float* __restrict__ next, int total4) {
    int i = blockIdx.x * blockDim.x + threadIdx.x;
    if (i < total4) {
        float sc = selfco[i >> 4];                   // 16 float4 per 64-wide row
        float4 v = ((const float4*)h)[i];
        v.x *= sc; v.y *= sc; v.z *= sc; v.w *= sc;
        ((float4*)next)[i] = v;
    }
}

// 16 threads per edge, float4 each: next[dst,:] += coef[e] * h[src,:]
// h (25.6 MB) lives in the 192 MB L2, so gathers + atomics are L2-resident.
__global__ void spmm_edges_kernel(const int* __restrict__ src, const int* __restrict__ dst,
                                  const float* __restrict__ coef,
                                  const float* __restrict__ h, float* __restrict__ next,
                                  int nE) {
    unsigned long long gid = (unsigned long long)blockIdx.x * blockDim.x + threadIdx.x;
    unsigned long long e = gid >> 4;
    if (e >= (unsigned long long)nE) return;
    const int c4 = (int)(gid & 15) << 2;             // col 0,4,...,60
    const int s = src[e];
    const int d = dst[e];
    const float w = coef[e];
    const float4 hv = *(const float4*)(h + (long)s * HID_DIM + c4);
    float* out = next + (long)d * HID_DIM + c4;
    atomic_add_f32(out + 0, w * hv.x);
    atomic_add_f32(out + 1, w * hv.y);
    atomic_add_f32(out + 2, w * hv.z);
    atomic_add_f32(out + 3, w * hv.w);
}

// =====================================================================
extern "C" void kernel_launch(void* const* d_in, const int* in_sizes, int n_in,
                              void* d_out, int out_size, void* d_ws, size_t ws_size,
                              hipStream_t stream) {
    const float* x     = (const float*)d_in[0];
    const int*   ei    = (const int*)  d_in[1];
    const float* emb_w = (const float*)d_in[2];
    const float* emb_b = (const float*)d_in[3];
    const float* ro_w  = (const float*)d_in[4];
    const float* ro_b  = (const float*)d_in[5];

    const int nN = in_sizes[0] / IN_DIM;             // 100000
    const int nE = in_sizes[1] / 2;                  // 1600000
    const int* srcIdx = ei;                          // edge_index[0,:]
    const int* dstIdx = ei + nE;                     // edge_index[1,:]

    // workspace layout (floats)
    float* ws     = (float*)d_ws;
    float* h0     = ws;                              // N*64
    float* h1     = h0 + (size_t)nN * HID_DIM;       // N*64
    float* dinv   = h1 + (size_t)nN * HID_DIM;       // N   (deg -> dinv in place)
    float* selfco = dinv + nN;                       // N
    float* coef   = selfco + nN;                     // E

    const int T = 256;
    const dim3 gGemm((nN + 127) / 128);
    const dim3 gNode((nN + T - 1) / T);
    const dim3 gEdge((nE + T - 1) / T);
    const int  total4 = nN * (HID_DIM / 4);
    const dim3 gScale((total4 + T - 1) / T);
    const unsigned long long spmmThreads = (unsigned long long)nE * 16ull;
    const dim3 gSpmm((unsigned)((spmmThreads + T - 1) / T));

    // 1) embed GEMM (WMMA f32, async-LDS weight staging + pair repack)
    gemm_embed_kernel<<<gGemm, T, 0, stream>>>(x, emb_w, emb_b, h0, nN);

    // 2) degree / norm precompute
    deg_init_kernel     <<<gNode, T, 0, stream>>>(dinv, nN);
    deg_edges_kernel    <<<gEdge, T, 0, stream>>>(dstIdx, dinv, nE);
    norm_finalize_kernel<<<gNode, T, 0, stream>>>(dinv, selfco, nN);
    coef_kernel         <<<gEdge, T, 0, stream>>>(srcIdx, dstIdx, dinv, coef, nE);

    // 3) 4 x (self-scale init + edge scatter-add), ping-pong h0<->h1
    float* hcur = h0;
    float* hnxt = h1;
    for (int it = 0; it < ITERATIONS; ++it) {
        scale_rows_kernel<<<gScale, T, 0, stream>>>(hcur, selfco, hnxt, total4);
        spmm_edges_kernel<<<gSpmm,  T, 0, stream>>>(srcIdx, dstIdx, coef, hcur, hnxt, nE);
        float* tmp = hcur; hcur = hnxt; hnxt = tmp;
    }
    // after 4 iterations hcur == h0

    // 4) tanh + readout GEMM (WMMA f32)
    gemm_readout_kernel<<<gGemm, T, 0, stream>>>(hcur, ro_w, ro_b, (float*)d_out, nN);
}